// MGN_50886772523302
// MI455X (gfx1250) — compile-verified
//
#include <hip/hip_runtime.h>

#define HIDDEN 128
#define PITCH  132                         // 128 + 4-dword pad: conflict-free b64 LDS reads
#define BSZ    (HIDDEN * PITCH)            // W region: 16896 dwords
#define AWSZ   (16 * PITCH)                // per-wave A strip: 2112 dwords
#define SMEMSZ (BSZ + 8 * AWSZ)            // 33792 dwords = 132KB (<= 320KB/WGP)

typedef __attribute__((ext_vector_type(2))) float v2f;
typedef __attribute__((ext_vector_type(4))) float v4f;
typedef __attribute__((ext_vector_type(8))) float v8f;

// ---------------------------------------------------------------------------
// Kernel 1: zero the workspace (agg + indeg), vectorized b128 stores.
// ---------------------------------------------------------------------------
__global__ __launch_bounds__(256) void MGN_zero_kernel(float* __restrict__ p, long n) {
  long i = (long)blockIdx.x * blockDim.x + threadIdx.x;
  long stride = (long)gridDim.x * blockDim.x;
  long n4 = n >> 2;
  for (long j = i; j < n4; j += stride)
    ((v4f*)p)[j] = (v4f){0.f, 0.f, 0.f, 0.f};
  for (long j = (n4 << 2) + i; j < n; j += stride)   // tail (none for this shape)
    p[j] = 0.f;
}

// ---------------------------------------------------------------------------
// Kernel 2: edge scatter-add (the runtime-dominant phase: 82M f32 L2 atomics).
// One wave32 per edge; lane l covers columns {l, l+32, l+64, l+96} so every
// load and every atomic is one fully coalesced 128B wave transaction.
// h (20.5MB) and agg (20.5MB) are L2-resident (192MB L2). The next edge's
// source row is prefetched (global_prefetch_b8) one wave-stride iteration
// ahead to hide L2 hit latency behind the current edge's atomics.
// ---------------------------------------------------------------------------
__global__ __launch_bounds__(256) void MGN_scatter_kernel(
    const float* __restrict__ h, const int* __restrict__ src,
    const int* __restrict__ dst, float* __restrict__ agg,
    float* __restrict__ indeg, int n_edges) {
  int lane = threadIdx.x & 31;
  int wid  = (int)((blockIdx.x * blockDim.x + threadIdx.x) >> 5);
  int nw   = (int)((gridDim.x * blockDim.x) >> 5);
  for (int e = wid; e < n_edges; e += nw) {
    int s = __builtin_amdgcn_readfirstlane(src[e]);
    int d = __builtin_amdgcn_readfirstlane(dst[e]);

    // prefetch next edge's source row (wave-uniform branch, 512B via 32 lanes)
    int en = e + nw;
    if (en < n_edges) {
      int sp = __builtin_amdgcn_readfirstlane(src[en]);
      const char* pf = (const char*)(h + (long)sp * HIDDEN);
      __builtin_prefetch(pf + lane * 16, 0, 0);
    }

    const float* hs = h   + (long)s * HIDDEN;
    float*       ad = agg + (long)d * HIDDEN;
#pragma unroll
    for (int k = 0; k < 4; ++k) {
      float v = hs[lane + 32 * k];
      atomicAdd(&ad[lane + 32 * k], v);       // global_atomic_add_f32, coalesced
    }
    if (lane == 0) atomicAdd(&indeg[d], 1.0f);
  }
}

// ---------------------------------------------------------------------------
// Kernel 3: updated = agg @ W^T + b via V_WMMA_F32_16X16X4_F32 (full fp32),
// fused with the indeg>0 select. One wave per 16-row strip; 8 accumulator
// tiles cover all 128 output columns.
//
// LDS layout (pitch-132 rows, all fragment reads bank-conflict-free):
//   W region : W[n][k] at dword n*132 + k           (67.6KB, staged by block)
//   A region : per-wave 16x128 strip, row r at dword r*132  (8.25KB x 8 waves)
// A fragment (lane lm=row, K pair per half) and B fragment (lane lm=col)
// addresses are base + k0*4 -> DS immediate offsets, no per-step VALU math.
// Per K-step: 1 + 8 ds_load_b64 into distinct regs, then 8 back-to-back WMMAs.
// ---------------------------------------------------------------------------
__global__ __launch_bounds__(256) void MGN_gemm_kernel(
    const float* __restrict__ agg, const float* __restrict__ W,
    const float* __restrict__ bias, const float* __restrict__ h,
    const float* __restrict__ indeg, float* __restrict__ out, int n_nodes) {
  __shared__ __align__(16) float smem[SMEMSZ];   // 132KB static LDS

  // ---- stage W (block-wide, coalesced b128 loads, conflict-free b128 stores)
  for (int i4 = threadIdx.x; i4 < (HIDDEN * HIDDEN) / 4; i4 += blockDim.x) {
    int idx = i4 << 2;                  // multiple of 4
    int n = idx >> 7, k = idx & 127;
    *(v4f*)&smem[n * PITCH + k] = *(const v4f*)&W[idx];
  }
  __syncthreads();

  int lane  = threadIdx.x & 31;
  int wave  = threadIdx.x >> 5;
  int strip = blockIdx.x * (blockDim.x >> 5) + wave;
  int row0  = strip << 4;
  if (row0 >= n_nodes) return;          // wave-uniform: EXEC stays all-1s

  int half = lane >> 4;                 // 0: K pair {0,1} / rows 0-7, 1: {2,3} / rows 8-15
  int lm   = lane & 15;

  // ---- stage this wave's 16x128 A strip (16 coalesced 512B global b128 loads)
  {
    const float* gsrc = agg + (long)row0 * HIDDEN;
    float* Aw = &smem[BSZ + wave * AWSZ];
#pragma unroll
    for (int r = 0; r < 16; ++r)
      *(v4f*)&Aw[r * PITCH + (lane << 2)] =
          *(const v4f*)&gsrc[r * HIDDEN + (lane << 2)];
  }

  // ---- fragment base addresses (DS immediate offsets carry k0 from here)
  const float* Abase = &smem[BSZ + wave * AWSZ + lm * PITCH + (half << 1)];
  const float* Bbase[8];
#pragma unroll
  for (int t = 0; t < 8; ++t)
    Bbase[t] = &smem[((t << 4) + lm) * PITCH + (half << 1)];

  v8f acc[8];
#pragma unroll
  for (int t = 0; t < 8; ++t)
    acc[t] = (v8f){0.f, 0.f, 0.f, 0.f, 0.f, 0.f, 0.f, 0.f};

#pragma unroll
  for (int k0 = 0; k0 < HIDDEN; k0 += 4) {
    v2f a = *(const v2f*)(Abase + k0);              // ds_load_b64, imm offset
    v2f bf[8];
#pragma unroll
    for (int t = 0; t < 8; ++t)
      bf[t] = *(const v2f*)(Bbase[t] + k0);         // ds_load_b64, imm offset
#pragma unroll
    for (int t = 0; t < 8; ++t)
      acc[t] = __builtin_amdgcn_wmma_f32_16x16x4_f32(
                   /*neg_a=*/false, a, /*neg_b=*/false, bf[t],
                   /*c_mod=*/(short)0, acc[t],
                   /*reuse_a=*/false, /*reuse_b=*/false);
  }

  // ---- epilogue: +bias, indeg>0 select, per C/D layout:
  // VGPR r holds row (row0 + r + 8*half), column = t*16 + lm.
#pragma unroll
  for (int t = 0; t < 8; ++t) {
    int col = (t << 4) + lm;
    float bc = bias[col];
#pragma unroll
    for (int r = 0; r < 8; ++r) {
      int m = row0 + r + (half << 3);
      long idx = (long)m * HIDDEN + col;
      float deg = indeg[m];
      out[idx] = (deg > 0.f) ? (acc[t][r] + bc) : h[idx];
    }
  }
}

// ---------------------------------------------------------------------------
extern "C" void kernel_launch(void* const* d_in, const int* in_sizes, int n_in,
                              void* d_out, int out_size, void* d_ws, size_t ws_size,
                              hipStream_t stream) {
  const float* h   = (const float*)d_in[0];   // [N, 128]
  const float* W   = (const float*)d_in[1];   // [128, 128]
  const float* b   = (const float*)d_in[2];   // [128]
  const int*   src = (const int*)d_in[3];     // [E]
  const int*   dst = (const int*)d_in[4];     // [E]
  float*       out = (float*)d_out;           // [N, 128]

  int n_nodes = in_sizes[0] / HIDDEN;
  int n_edges = in_sizes[3];

  float* agg   = (float*)d_ws;                        // [N, 128]
  float* indeg = agg + (long)n_nodes * HIDDEN;        // [N]
  long   zn    = (long)n_nodes * HIDDEN + n_nodes;

  MGN_zero_kernel<<<512, 256, 0, stream>>>(agg, zn);
  MGN_scatter_kernel<<<2048, 256, 0, stream>>>(h, src, dst, agg, indeg, n_edges);

  int strips = (n_nodes + 15) / 16;
  int blocks = (strips + 7) / 8;                      // 8 waves per 256-thread block
  MGN_gemm_kernel<<<blocks, 256, 0, stream>>>(agg, W, b, h, indeg, out, n_nodes);
}